// CenterNet_20950850470032
// MI455X (gfx1250) — compile-verified
//
#include <hip/hip_runtime.h>
#include <hip/hip_bf16.h>
#include <stdint.h>

// ---------------------------------------------------------------------------
// CDNA5 (gfx1250) async global->LDS copy primitives (inline asm; portable
// across ROCm7.2 / amdgpu-toolchain since it bypasses the clang builtins).
// ---------------------------------------------------------------------------
__device__ __forceinline__ void async_copy_b128(unsigned lds_addr, const void* gptr) {
    asm volatile("global_load_async_to_lds_b128 %0, %1, off"
                 :: "v"(lds_addr), "v"((unsigned long long)(uintptr_t)gptr)
                 : "memory");
}
__device__ __forceinline__ void wait_async0() {
    asm volatile("s_wait_asynccnt 0" ::: "memory");
}

// ---------------------------------------------------------------------------
// Tensor Data Mover: 2D tile load (global -> LDS), one instruction per tile.
// D# built in SGPRs: group0 (4 dw) + group1 (8 dw); VADDR2/3 = NULL (<=2D).
// Tracked by TENSORcnt.  (CDNA5 ISA: CDNA5 ISA/§8)
// ---------------------------------------------------------------------------
typedef unsigned int u32x4 __attribute__((ext_vector_type(4)));
typedef unsigned int u32x8 __attribute__((ext_vector_type(8)));

__device__ __forceinline__ void tdm_load_2d_f32(unsigned lds_addr, const void* gptr,
                                                unsigned tile_w, unsigned tile_h,
                                                unsigned row_stride_elems) {
    const unsigned long long ga = (unsigned long long)(uintptr_t)gptr;
    u32x4 g0;
    g0.x = 1u;                                              // count=1 (valid user D#)
    g0.y = lds_addr;                                        // lds_addr[31:0]
    g0.z = (unsigned)ga;                                    // global_addr[31:0]
    g0.w = (unsigned)((ga >> 32) & 0x01FFFFFFu) | (2u << 30); // addr[56:32] | type=2

    u32x8 g1;
    g1.s0 = 2u << 16;                                       // wg_mask=0, data_size=4B
    g1.s1 = (tile_w & 0xFFFFu) << 16;                       // tensor_dim0[15:0]
    g1.s2 = ((tile_w >> 16) & 0xFFFFu) | ((tile_h & 0xFFFFu) << 16);   // dim0 hi | dim1 lo
    g1.s3 = ((tile_h >> 16) & 0xFFFFu) | ((tile_w & 0xFFFFu) << 16);   // dim1 hi | tile_dim0
    g1.s4 = tile_h & 0xFFFFu;                               // tile_dim1 ; tile_dim2=0
    g1.s5 = row_stride_elems;                               // tensor_dim0_stride[31:0]
    g1.s6 = 0u;                                             // stride hi | dim1_stride
    g1.s7 = 0u;
    asm volatile("tensor_load_to_lds %0, %1" :: "s"(g0), "s"(g1) : "memory");
}
// s_wait_tensorcnt takes an immediate -> dedicated helpers with literal args.
__device__ __forceinline__ void wait_tensor0() {
    __builtin_amdgcn_s_wait_tensorcnt(0);
}
__device__ __forceinline__ void wait_tensor1() {
    __builtin_amdgcn_s_wait_tensorcnt(1);
}

// Low 32 bits of a flat shared-aperture address are the LDS byte offset
// (CDNA5 ISA 10.2).
__device__ __forceinline__ unsigned lds_off(const void* p) {
    return (unsigned)(uintptr_t)p;
}

// Block-wide max of packed u64 keys (256 threads).
__device__ __forceinline__ unsigned long long
block_max_u64(unsigned long long key, unsigned long long* red, int tid) {
    red[tid] = key;
    __syncthreads();
    #pragma unroll
    for (int s = 128; s > 0; s >>= 1) {
        if (tid < s) {
            unsigned long long o = red[tid + s];
            if (o > red[tid]) red[tid] = o;
        }
        __syncthreads();
    }
    unsigned long long r = red[0];
    __syncthreads();
    return r;
}

#define NCLS  80
#define HW    65536   // 256*256
#define TOPK  100
#define CAP   4096    // peak-candidate capacity per class (expected ~2600)

// ---------------------------------------------------------------------------
// Kernel A: per-class  sigmoid + 5x5 peak-keep + top-100.
// One block per class (80 blocks, 256 threads = 8 wave32).
// TDM streams each 12x256 strip (8 rows + 2 halo) into LDS; DOUBLE-BUFFERED:
// strip s+1's DMA is issued before computing strip s, so the HBM round-trip
// latency of every transfer is hidden behind compute (TDM ops of one wave
// complete in order, so s_wait_tensorcnt<=1 guarantees strip s has landed).
// Peak test runs on logits (sigmoid is monotonic); sigmoid applied to peaks.
// ---------------------------------------------------------------------------
__global__ __launch_bounds__(256) void
peak_topk_per_class(const float* __restrict__ cls,
                    float* __restrict__ ws_scores,   // [80*100]
                    int*   __restrict__ ws_inds) {   // [80*100]
    __shared__ __align__(16) float tile[2][12 * 256];   // ping-pong strip buffers
    __shared__ float cand_val[CAP];
    __shared__ int   cand_idx[CAP];
    __shared__ unsigned long long red[256];
    __shared__ int cnt;

    const int tid = threadIdx.x;
    const int c   = blockIdx.x;
    const float* base = cls + (size_t)c * HW;        // batch 0, class c

    if (tid == 0) cnt = 0;

    const unsigned tbase0 = lds_off(&tile[0][0]);
    const unsigned tbase1 = lds_off(&tile[1][0]);

    // Prologue: kick off strip 0's DMA (wave 0 only; TDM ignores EXEC, so
    // issuing from every wave would duplicate the transfer 8x).
    if (tid < 32) {
        // strip 0: rows [0, 10) land at LDS rows 2..11 (rows 0..1 unused halo)
        tdm_load_2d_f32(tbase0 + 2u * 256u * 4u, base,
                        /*tile_w=*/256, /*tile_h=*/10, /*row_stride=*/256);
    }
    __syncthreads();

    for (int strip = 0; strip < 32; ++strip) {
        const int y0 = strip * 8;

        if (tid < 32) {
            if (strip + 1 < 32) {
                // Issue next strip's DMA into the other buffer, then wait for
                // the *previous* (oldest) transfer only: TENSORcnt <= 1.
                const int ny0 = y0 + 8;
                int r0 = ny0 - 2;
                int r1 = ny0 + 10 > 256 ? 256 : ny0 + 10;
                const unsigned nxt = ((strip + 1) & 1) ? tbase1 : tbase0;
                tdm_load_2d_f32(nxt, base + r0 * 256,
                                /*tile_w=*/256, /*tile_h=*/(unsigned)(r1 - r0),
                                /*row_stride=*/256);
                wait_tensor1();
            } else {
                wait_tensor0();      // drain: last strip's DMA must be done
            }
        }
        __syncthreads();             // current tile visible to all waves

        const float* t = (strip & 1) ? tile[1] : tile[0];
        const int x = tid;           // one column per thread, 8 rows per strip
        #pragma unroll
        for (int j = 0; j < 8; ++j) {
            const int y = y0 + j;
            const float center = t[(j + 2) * 256 + x];
            float m = center;
            #pragma unroll
            for (int dy = -2; dy <= 2; ++dy) {
                const int yy = y + dy;
                if (yy < 0 || yy > 255) continue;
                const int rr = (j + 2 + dy) * 256;
                #pragma unroll
                for (int dx = -2; dx <= 2; ++dx) {
                    const int xx = x + dx;
                    if (xx < 0 || xx > 255) continue;
                    const float v = t[rr + xx];
                    m = v > m ? v : m;
                }
            }
            if (center >= m) {                        // local 5x5 maximum
                const float score = 1.0f / (1.0f + __expf(-center));
                const int p = atomicAdd(&cnt, 1);
                if (p < CAP) { cand_val[p] = score; cand_idx[p] = y * 256 + x; }
            }
        }
        __syncthreads();             // done reading this buffer before reuse
    }

    const int n = cnt < CAP ? cnt : CAP;

    // 100 x block-wide arg-max extraction. Key packs (f32 bits, ~pos): max key
    // = (highest score, lowest position); positive f32 bits are monotonic.
    for (int k = 0; k < TOPK; ++k) {
        unsigned long long lk = 0;
        for (int p = tid; p < n; p += 256) {
            const float v = cand_val[p];
            if (v >= 0.0f) {
                const unsigned long long key =
                    ((unsigned long long)__float_as_uint(v) << 32) |
                    (unsigned long long)(0xFFFFFFFFu - (unsigned)p);
                if (key > lk) lk = key;
            }
        }
        const unsigned long long best = block_max_u64(lk, red, tid);
        if (tid == 0) {
            if (best) {
                const unsigned p = 0xFFFFFFFFu - (unsigned)(best & 0xFFFFFFFFull);
                ws_scores[c * TOPK + k] = __uint_as_float((unsigned)(best >> 32));
                ws_inds[c * TOPK + k]   = cand_idx[p];
                cand_val[p] = -1.0f;                  // remove winner
            } else {                                   // <100 peaks: pad
                ws_scores[c * TOPK + k] = 0.0f;
                ws_inds[c * TOPK + k]   = 0;
            }
        }
        __syncthreads();
    }
}

// ---------------------------------------------------------------------------
// Kernel B: global top-100 over the 8000 class-major candidates + box decode.
// Single block, 256 threads. Candidate position p -> class = p/100 (matches
// the reference's i2 // TOPK), spatial index from ws_inds[p].
// ---------------------------------------------------------------------------
__global__ __launch_bounds__(256) void
global_topk_decode(const float* __restrict__ ws_scores,
                   const int*   __restrict__ ws_inds,
                   const float* __restrict__ txty,   // batch 0: [2,256,256]
                   const float* __restrict__ twth,   // batch 0: [2,256,256]
                   float* __restrict__ out) {        // 600 floats
    __shared__ __align__(16) float vals[NCLS * TOPK];   // 8000
    __shared__ unsigned long long red[256];
    __shared__ int   winpos[TOPK];
    __shared__ float winval[TOPK];

    const int tid = threadIdx.x;
    const unsigned vbase = lds_off(&vals[0]);

    // Async-copy the 8000 scores into LDS (2000 x b128, ASYNCcnt-tracked).
    for (int i = tid; i < 2000; i += 256)
        async_copy_b128(vbase + (unsigned)(i * 16), ws_scores + i * 4);
    wait_async0();
    __syncthreads();

    for (int k = 0; k < TOPK; ++k) {
        unsigned long long lk = 0;
        for (int p = tid; p < NCLS * TOPK; p += 256) {
            const float v = vals[p];
            if (v >= 0.0f) {
                const unsigned long long key =
                    ((unsigned long long)__float_as_uint(v) << 32) |
                    (unsigned long long)(0xFFFFFFFFu - (unsigned)p);
                if (key > lk) lk = key;
            }
        }
        const unsigned long long best = block_max_u64(lk, red, tid);
        if (tid == 0) {
            const unsigned p = 0xFFFFFFFFu - (unsigned)(best & 0xFFFFFFFFull);
            winpos[k] = (int)p;
            winval[k] = __uint_as_float((unsigned)(best >> 32));
            vals[p]   = -1.0f;
        }
        __syncthreads();
    }

    // Decode the 100 winning boxes in parallel.
    if (tid < TOPK) {
        const int p   = winpos[tid];
        const int cid = p / TOPK;
        const int s   = ws_inds[p];
        const int y = s >> 8, x = s & 255;

        const float tx = txty[s];
        const float ty = txty[HW + s];
        const float tw = twth[s];
        const float th = twth[HW + s];

        const float cx = (1.0f / (1.0f + __expf(-tx)) + (float)x) * 4.0f; // STRIDE=4
        const float cy = (1.0f / (1.0f + __expf(-ty)) + (float)y) * 4.0f;
        const float bw = __expf(tw) * 4.0f;
        const float bh = __expf(th) * 4.0f;

        const float inv = 1.0f / 1024.0f;   // H*STRIDE
        float x0 = (cx - 0.5f * bw) * inv, y0 = (cy - 0.5f * bh) * inv;
        float x1 = (cx + 0.5f * bw) * inv, y1 = (cy + 0.5f * bh) * inv;
        x0 = fminf(fmaxf(x0, 0.0f), 1.0f);  y0 = fminf(fmaxf(y0, 0.0f), 1.0f);
        x1 = fminf(fmaxf(x1, 0.0f), 1.0f);  y1 = fminf(fmaxf(y1, 0.0f), 1.0f);

        out[tid * 4 + 0] = x0;
        out[tid * 4 + 1] = y0;
        out[tid * 4 + 2] = x1;
        out[tid * 4 + 3] = y1;
        out[400 + tid] = winval[tid];
        out[500 + tid] = (float)cid;        // class id as float in float d_out
    }
}

// ---------------------------------------------------------------------------
extern "C" void kernel_launch(void* const* d_in, const int* in_sizes, int n_in,
                              void* d_out, int out_size, void* d_ws, size_t ws_size,
                              hipStream_t stream) {
    const float* cls  = (const float*)d_in[0];  // [8,80,256,256] - only batch 0 used
    const float* txty = (const float*)d_in[1];  // [8,2,256,256]  - only batch 0 used
    const float* twth = (const float*)d_in[2];  // [8,2,256,256]  - only batch 0 used
    float* out = (float*)d_out;                 // 400 bbox + 100 scores + 100 clses

    float* ws_scores = (float*)d_ws;                                      // 8000 f32
    int*   ws_inds   = (int*)((char*)d_ws + NCLS * TOPK * sizeof(float)); // 8000 i32

    peak_topk_per_class<<<NCLS, 256, 0, stream>>>(cls, ws_scores, ws_inds);
    global_topk_decode<<<1, 256, 0, stream>>>(ws_scores, ws_inds, txty, twth, out);
}